// Uiqi_pan_80874234184239
// MI455X (gfx1250) — compile-verified
//
#include <hip/hip_runtime.h>
#include <hip/hip_bf16.h>

#define EPS 1e-8f

typedef float v2f __attribute__((ext_vector_type(2)));
typedef float v8f __attribute__((ext_vector_type(8)));

// D = A(16x4) * B(4x16) + C, f32 WMMA (CDNA5 V_WMMA_F32_16X16X4_F32).
// 8-arg pattern (probe-confirmed for 16x16x{4,32} family):
// (neg_a, A, neg_b, B, c_mod, C, reuse_a, reuse_b)
__device__ __forceinline__ v8f wmma4(v2f a, v2f b, v8f c) {
    return __builtin_amdgcn_wmma_f32_16x16x4_f32(false, a, false, b, (short)0, c,
                                                 false, false);
}

// ---------------------------------------------------------------------------
// Kernel 1: per-block partial sums of pan, pan^2, img1 (deterministic tree).
// 2048 blocks x 256 threads x 8 elems = 4,194,304 = 16*512*512 exactly.
// ---------------------------------------------------------------------------
__global__ __launch_bounds__(256) void uiqi_reduce1(const float* __restrict__ img1,
                                                    const float* __restrict__ pan,
                                                    float* __restrict__ partials) {
    __shared__ float r0[256], r1[256], r2[256];
    const int tid = threadIdx.x;
    const size_t base = (size_t)blockIdx.x * 2048 + tid;
    float sp = 0.f, spp = 0.f, si = 0.f;
#pragma unroll
    for (int j = 0; j < 8; ++j) {
        const size_t idx = base + (size_t)j * 256;
        const float pv = pan[idx];
        const float iv = img1[idx];
        sp += pv;
        spp += pv * pv;
        si += iv;
    }
    r0[tid] = sp; r1[tid] = spp; r2[tid] = si;
    __syncthreads();
    for (int s = 128; s > 0; s >>= 1) {
        if (tid < s) { r0[tid] += r0[tid + s]; r1[tid] += r1[tid + s]; r2[tid] += r2[tid + s]; }
        __syncthreads();
    }
    if (tid == 0) {
        partials[blockIdx.x * 3 + 0] = r0[0];
        partials[blockIdx.x * 3 + 1] = r1[0];
        partials[blockIdx.x * 3 + 2] = r2[0];
    }
}

// ---------------------------------------------------------------------------
// Kernel 2: fold 2048 partials -> scalars {mean_pan, inv_std_pan, mean_img1}.
// std is unbiased (ddof=1), matching torch/jnp.std(ddof=1).
// Note: the reference's second normalization p*std(p,ddof=1)+mean(img1) is
// exactly p + mean(img1) since std((x-mu)/sigma, ddof=1) == 1.
// ---------------------------------------------------------------------------
__global__ __launch_bounds__(256) void uiqi_reduce2(const float* __restrict__ partials,
                                                    float* __restrict__ sc) {
    __shared__ float r0[256], r1[256], r2[256];
    const int tid = threadIdx.x;
    float sp = 0.f, spp = 0.f, si = 0.f;
    for (int i = tid; i < 2048; i += 256) {
        sp += partials[3 * i + 0];
        spp += partials[3 * i + 1];
        si += partials[3 * i + 2];
    }
    r0[tid] = sp; r1[tid] = spp; r2[tid] = si;
    __syncthreads();
    for (int s = 128; s > 0; s >>= 1) {
        if (tid < s) { r0[tid] += r0[tid + s]; r1[tid] += r1[tid + s]; r2[tid] += r2[tid + s]; }
        __syncthreads();
    }
    if (tid == 0) {
        const double N = 4194304.0;  // 16*512*512
        const double S = (double)r0[0], SS = (double)r1[0];
        const double mean = S / N;
        const double var = (SS - S * S / N) / (N - 1.0);
        sc[0] = (float)mean;
        sc[1] = (float)(1.0 / sqrt(var));
        sc[2] = (float)((double)r2[0] / N);
    }
}

// ---------------------------------------------------------------------------
// Main kernel: one block per (col-tile, image).
// Phase 1: horizontal 32-tap circular box sums (sliding window) into LDS H[512][16]
//          for fields a=img1, p=(pan-mu)*inv_std+mean(img1), ab=a*p.
// Phase 2: vertical 32-tap sums as banded-ones GEMM on WMMA f32 16x16x4:
//          S(16 out-rows x 16 cols) = W(16x47 band) x H(47x16), 12 K=4 chunks.
// Epilogue: UIQI q per element, masked 513x513, deterministic block reduction.
// ---------------------------------------------------------------------------
__global__ __launch_bounds__(256) void uiqi_main(const float* __restrict__ img1,
                                                 const float* __restrict__ pan,
                                                 const float* __restrict__ sc,
                                                 float* __restrict__ Qpart) {
    __shared__ float Ha[512 * 16];
    __shared__ float Hp[512 * 16];
    __shared__ float Hab[512 * 16];
    __shared__ float red[256];

    const int tid = threadIdx.x;
    const int ct = blockIdx.x;   // 0..32 column tile (16 output cols each)
    const int img = blockIdx.y;  // 0..15
    const float mp = sc[0], invs = sc[1], mi1 = sc[2];
    const int c0 = ct * 16;
    const int cb = c0 - 16;  // first column of the leftmost window
    const float* __restrict__ Aimg = img1 + (size_t)img * 512 * 512;
    const float* __restrict__ Pimg = pan + (size_t)img * 512 * 512;

    // ---- Phase 1: sliding horizontal sums (2 rows per thread) ----
    for (int r = tid; r < 512; r += 256) {
        const float* __restrict__ ar = Aimg + (size_t)r * 512;
        const float* __restrict__ pr = Pimg + (size_t)r * 512;
        float sa = 0.f, sp = 0.f, sab = 0.f;
#pragma unroll
        for (int j = 0; j < 32; ++j) {
            const int c = (cb + j) & 511;  // circular pad
            const float av = ar[c];
            const float pv = (pr[c] - mp) * invs + mi1;
            sa += av; sp += pv; sab += av * pv;
        }
        Ha[r * 16] = sa; Hp[r * 16] = sp; Hab[r * 16] = sab;
#pragma unroll
        for (int j = 1; j < 16; ++j) {
            const int ca = (cb + 31 + j) & 511;
            const int cs = (cb + j - 1) & 511;
            const float av1 = ar[ca], av0 = ar[cs];
            const float pv1 = (pr[ca] - mp) * invs + mi1;
            const float pv0 = (pr[cs] - mp) * invs + mi1;
            sa += av1 - av0;
            sp += pv1 - pv0;
            sab += av1 * pv1 - av0 * pv0;
            Ha[r * 16 + j] = sa; Hp[r * 16 + j] = sp; Hab[r * 16 + j] = sab;
        }
    }
    __syncthreads();

    // ---- Phase 2: vertical 32-tap sums via WMMA ----
    const int lane = tid & 31;
    const int wave = tid >> 5;
    const int n = lane & 15;    // output column within tile / A-matrix row m
    const int half = lane >> 4; // lane half selects K pairs (ISA A/B layout)

    // A operand = band matrix W[m][k] = (m <= k <= m+31); per-lane constant.
    // ISA f32 A 16x4 layout: lanes0-15 hold K={0,1} in vgpr{0,1}; lanes16-31 K={2,3}.
    v2f aw[12];
#pragma unroll
    for (int kk = 0; kk < 12; ++kk) {
        const int k0 = 4 * kk + 2 * half;
        aw[kk].x = (k0 >= n && k0 <= n + 31) ? 1.0f : 0.0f;
        aw[kk].y = (k0 + 1 >= n && k0 + 1 <= n + 31) ? 1.0f : 0.0f;
    }

    float qsum = 0.0f;
    const float inv = 1.0f / 1024.0f;

    for (int tile = wave; tile < 33; tile += 8) {
        const int base = tile * 16 - 16;  // first strip row (k=0); &511 handles wrap
        v8f s1 = {}, s2 = {}, s12 = {};
#pragma unroll
        for (int kk = 0; kk < 12; ++kk) {
            const int k0 = 4 * kk + 2 * half;
            const int r0 = (base + k0) & 511;
            const int r1 = (base + k0 + 1) & 511;
            v2f b1, b2, b12;
            b1.x = Ha[r0 * 16 + n];  b1.y = Ha[r1 * 16 + n];
            b2.x = Hp[r0 * 16 + n];  b2.y = Hp[r1 * 16 + n];
            b12.x = Hab[r0 * 16 + n]; b12.y = Hab[r1 * 16 + n];
            s1 = wmma4(aw[kk], b1, s1);
            s2 = wmma4(aw[kk], b2, s2);
            s12 = wmma4(aw[kk], b12, s12);
        }
        // Epilogue: UIQI q per element (branchless; EXEC stays full around WMMA).
#pragma unroll
        for (int g = 0; g < 8; ++g) {
            const float mu1 = s1[g] * inv;
            const float mu2 = s2[g] * inv;
            const float mu1sq = mu1 * mu1, mu2sq = mu2 * mu2, mu12 = mu1 * mu2;
            const float sg1 = mu1 - mu1sq;  // faithful: source uses avg(img1)-mu^2
            const float sg2 = mu2 - mu2sq;
            const float sg12 = s12[g] * inv - mu12;
            const float s = sg1 + sg2;
            const float m_ = mu1sq + mu2sq;
            const float ss = (__builtin_fabsf(s) > 0.0f) ? s : 1.0f;
            const float sm = (__builtin_fabsf(m_) > 0.0f) ? m_ : 1.0f;
            float q = 1.0f;
            q = (s < EPS && m_ > EPS) ? (2.0f * mu12 / sm) : q;
            q = (s > EPS && m_ < EPS) ? (2.0f * sg12 / ss) : q;
            q = (s > EPS && m_ > EPS) ? (4.0f * mu12 * sg12 / (sm * ss)) : q;
            const int orow = tile * 16 + g + 8 * half;  // C/D layout: M = vgpr + 8*half
            const int ocol = c0 + n;
            qsum += (orow < 513 && ocol < 513) ? q : 0.0f;
        }
    }

    // Deterministic block reduction of q partials.
    red[tid] = qsum;
    __syncthreads();
    for (int st = 128; st > 0; st >>= 1) {
        if (tid < st) red[tid] += red[tid + st];
        __syncthreads();
    }
    if (tid == 0) Qpart[img * 33 + ct] = red[0];
}

// ---------------------------------------------------------------------------
// Finalize: fixed-order sum of the 528 block partials, divide by 16*513*513.
// ---------------------------------------------------------------------------
__global__ __launch_bounds__(256) void uiqi_finalize(const float* __restrict__ Qpart,
                                                     float* __restrict__ out) {
    __shared__ float red[256];
    const int tid = threadIdx.x;
    float s = 0.f;
    for (int i = tid; i < 528; i += 256) s += Qpart[i];
    red[tid] = s;
    __syncthreads();
    for (int st = 128; st > 0; st >>= 1) {
        if (tid < st) red[tid] += red[tid + st];
        __syncthreads();
    }
    if (tid == 0) out[0] = red[0] / (16.0f * 513.0f * 513.0f);
}

extern "C" void kernel_launch(void* const* d_in, const int* in_sizes, int n_in,
                              void* d_out, int out_size, void* d_ws, size_t ws_size,
                              hipStream_t stream) {
    (void)in_sizes; (void)n_in; (void)out_size; (void)ws_size;
    const float* img1 = (const float*)d_in[0];
    const float* pan = (const float*)d_in[1];
    float* ws = (float*)d_ws;
    float* sc = ws;                       // 4 floats: mean_pan, inv_std, mean_img1
    float* partials = ws + 4;             // 2048 * 3 floats
    float* Qpart = ws + 4 + 2048 * 3;     // 528 floats
    float* out = (float*)d_out;

    uiqi_reduce1<<<2048, 256, 0, stream>>>(img1, pan, partials);
    uiqi_reduce2<<<1, 256, 0, stream>>>(partials, sc);
    uiqi_main<<<dim3(33, 16, 1), 256, 0, stream>>>(img1, pan, sc, Qpart);
    uiqi_finalize<<<1, 256, 0, stream>>>(Qpart, out);
}